// GCN_21320217657536
// MI455X (gfx1250) — compile-verified
//
#include <hip/hip_runtime.h>

// GCN: 4-hop propagation + per-graph dense layer on MI455X (gfx1250).
// SpMM hops: wave32-per-edge float4 gather + global_atomic_add_f32 scatter
// (whole working set is L2-resident on MI455X's 192MB L2).
// Dense layer: V_WMMA_F32_16X16X4_F32 (exact fp32, matches reference matmul).

typedef __attribute__((ext_vector_type(2))) float v2f;
typedef __attribute__((ext_vector_type(8))) float v8f;

#define DFEAT 128
#define BETA 0.1f
#define NUM_HOP 4

__device__ __forceinline__ void atomicAddF(float* p, float v) {
  unsafeAtomicAdd(p, v);   // lowers to global_atomic_add_f32 on gfx1250
}

// ---- in-degree via float atomics -------------------------------------------
__global__ __launch_bounds__(256) void deg_kernel(const int* __restrict__ dst,
                                                  float* __restrict__ deg, int E) {
  int i = blockIdx.x * blockDim.x + threadIdx.x;
  if (i < E) atomicAddF(&deg[dst[i]], 1.0f);
}

__global__ __launch_bounds__(256) void norm_kernel(float* __restrict__ norm, int N) {
  int i = blockIdx.x * blockDim.x + threadIdx.x;
  if (i < N) {
    float d = norm[i];
    d = d < 1.0f ? 1.0f : d;
    norm[i] = 1.0f / sqrtf(d);   // clip(deg,1)^-0.5
  }
}

// ---- one hop: agg[dst] += h[src] * norm[src] * w[e] ------------------------
// One wave32 per edge; lane l owns features [4l, 4l+3] (128/32 = float4/lane).
__global__ __launch_bounds__(256) void edge_scatter(
    const int* __restrict__ src, const int* __restrict__ dst,
    const float* __restrict__ ef, const float* __restrict__ norm,
    const float* __restrict__ h, float* __restrict__ agg, int E) {
  int e = (blockIdx.x * blockDim.x + threadIdx.x) >> 5;
  int lane = threadIdx.x & 31;
  if (e >= E) return;
  int s = src[e];
  int d = dst[e];
  float scale = norm[s] * ef[e];
  const float4* hrow = (const float4*)(h + (size_t)s * DFEAT);
  float4 v = hrow[lane];
  float* arow = agg + (size_t)d * DFEAT + lane * 4;
  atomicAddF(arow + 0, v.x * scale);
  atomicAddF(arow + 1, v.y * scale);
  atomicAddF(arow + 2, v.z * scale);
  atomicAddF(arow + 3, v.w * scale);
}

// ---- residual mix: h = 0.9*agg + 0.1*h0 ------------------------------------
__global__ __launch_bounds__(256) void mix_kernel(const float4* __restrict__ agg,
                                                  const float4* __restrict__ h0,
                                                  float4* __restrict__ h, int n4) {
  int i = blockIdx.x * blockDim.x + threadIdx.x;
  if (i < n4) {
    float4 a = agg[i], f = h0[i], r;
    r.x = a.x * (1.0f - BETA) + f.x * BETA;
    r.y = a.y * (1.0f - BETA) + f.y * BETA;
    r.z = a.z * (1.0f - BETA) + f.z * BETA;
    r.w = a.w * (1.0f - BETA) + f.w * BETA;
    h[i] = r;
  }
}

// ---- out[:, colBase+16*colTile ...] = relu(h @ W + b) via fp32 WMMA --------
// One wave per 16x16 tile. 32x V_WMMA_F32_16X16X4_F32 along K=128.
// No lane-divergent exits before WMMA (EXEC must stay all-1s).
__global__ __launch_bounds__(32) void gemm_relu_wmma(
    const float* __restrict__ h,     // [N,128]
    const float* __restrict__ W,     // [128,128] row-major (k,n)
    const float* __restrict__ bias,  // [128]
    float* __restrict__ out,         // [N, ostride]
    int N, int colBase, int ostride) {
  int rowTile = blockIdx.x;
  int colTile = blockIdx.y;
  int lane = threadIdx.x;
  int half = lane >> 4;   // K-group selector for A/B, row-group for C/D
  int ln   = lane & 15;   // A row / B,C,D column within tile

  int m = rowTile * 16 + ln;
  int mload = m < N ? m : N - 1;           // clamp, keep EXEC uniform
  int n = colTile * 16 + ln;

  const float* arow = h + (size_t)mload * DFEAT;
  const float* bcol = W + n;

  v8f acc = {};
#pragma unroll
  for (int k0 = 0; k0 < DFEAT; k0 += 4) {
    int k = k0 + 2 * half;
    v2f a;  a.x = arow[k];                  a.y = arow[k + 1];
    v2f b;  b.x = bcol[(size_t)k * DFEAT];  b.y = bcol[(size_t)(k + 1) * DFEAT];
    acc = __builtin_amdgcn_wmma_f32_16x16x4_f32(
        /*neg_a=*/false, a, /*neg_b=*/false, b,
        /*c_mod=*/(short)0, acc, /*reuse_a=*/false, /*reuse_b=*/false);
  }

  float bv = bias[n];
#pragma unroll
  for (int v = 0; v < 8; ++v) {
    int mrow = rowTile * 16 + half * 8 + v;   // C/D layout: row = v + 8*half
    if (mrow < N) {
      float r = acc[v] + bv;
      out[(size_t)mrow * ostride + colBase + n] = r > 0.0f ? r : 0.0f;
    }
  }
}

extern "C" void kernel_launch(void* const* d_in, const int* in_sizes, int n_in,
                              void* d_out, int out_size, void* d_ws, size_t ws_size,
                              hipStream_t stream) {
  const float* features = (const float*)d_in[0];  // [N,128]
  const int*   src      = (const int*)d_in[1];    // [E]
  const int*   dstv     = (const int*)d_in[2];    // [E]
  const float* ef       = (const float*)d_in[3];  // [G,E]
  const float* W        = (const float*)d_in[4];  // [G,128,128]
  const float* b        = (const float*)d_in[5];  // [G,128]
  float* out = (float*)d_out;

  const int D = DFEAT;
  const int N = in_sizes[0] / D;
  const int E = in_sizes[1];
  const int G = in_sizes[3] / E;
  const int ostride = G * D;

  // Workspace: norm[N] | h[N*D] | agg[N*D]   (~51.4 MB for reference sizes)
  float* norm = (float*)d_ws;
  size_t normElems = ((size_t)N + 3) & ~(size_t)3;   // keep float4 alignment
  float* h   = norm + normElems;
  float* agg = h + (size_t)N * D;

  // degrees -> norm
  hipMemsetAsync(norm, 0, (size_t)N * sizeof(float), stream);
  deg_kernel<<<(E + 255) / 256, 256, 0, stream>>>(dstv, norm, E);
  norm_kernel<<<(N + 255) / 256, 256, 0, stream>>>(norm, N);

  const int n4 = (N * D) / 4;
  const int scatterBlocks = (int)(((long long)E * 32 + 255) / 256);

  for (int g = 0; g < G; ++g) {
    hipMemcpyAsync(h, features, (size_t)N * D * sizeof(float),
                   hipMemcpyDeviceToDevice, stream);
    const float* efg = ef + (size_t)g * E;
    for (int hop = 0; hop < NUM_HOP; ++hop) {
      hipMemsetAsync(agg, 0, (size_t)N * D * sizeof(float), stream);
      edge_scatter<<<scatterBlocks, 256, 0, stream>>>(src, dstv, efg, norm, h,
                                                      agg, E);
      mix_kernel<<<(n4 + 255) / 256, 256, 0, stream>>>(
          (const float4*)agg, (const float4*)features, (float4*)h, n4);
    }
    dim3 grid((N + 15) / 16, D / 16);
    gemm_relu_wmma<<<grid, 32, 0, stream>>>(h, W + (size_t)g * D * D,
                                            b + (size_t)g * D, out, N, g * D,
                                            ostride);
  }
}